// CrossAttention_45724221833727
// MI455X (gfx1250) — compile-verified
//
#include <hip/hip_runtime.h>

#define B_ 8
#define S_ 2048
#define D_ 512
#define M_ (B_ * S_)

typedef float v2f __attribute__((ext_vector_type(2)));
typedef float v8f __attribute__((ext_vector_type(8)));
typedef unsigned int u32x4 __attribute__((ext_vector_type(4)));
typedef int i32x4 __attribute__((ext_vector_type(4)));
typedef int i32x8 __attribute__((ext_vector_type(8)));

// Native fp32 WMMA: D(16x16,f32) = A(16x4,f32) x B(4x16,f32) + C
__device__ __forceinline__ v8f wmma_f32(v2f a, v2f b, v8f c) {
  return __builtin_amdgcn_wmma_f32_16x16x4_f32(false, a, false, b, (short)0, c,
                                               false, false);
}

// ---------------------------------------------------------------------------
// Tensor Data Mover: 2D tile load Global -> LDS (D# per CDNA5 ISA)
// ---------------------------------------------------------------------------
__device__ __forceinline__ void tdm_load_2d(uint32_t lds_byte_addr,
                                            const void* gptr, uint32_t tile0,
                                            uint32_t tile1, uint32_t tdim0,
                                            uint32_t tdim1, uint64_t stride0,
                                            uint32_t flags) {
  const uint64_t ga = (uint64_t)(uintptr_t)gptr;
  u32x4 g0;
  g0.x = 1u;  // count=1 (valid user descriptor)
  g0.y = lds_byte_addr;
  g0.z = (uint32_t)(ga & 0xffffffffu);
  g0.w = (uint32_t)((ga >> 32) & 0x01ffffffu) | (2u << 30);  // type=2 image
  i32x8 g1;
  g1[0] = (int)flags;  // data_size + pad controls
  g1[1] = (int)((tdim0 & 0xffffu) << 16);
  g1[2] = (int)((tdim0 >> 16) | ((tdim1 & 0xffffu) << 16));
  g1[3] = (int)((tdim1 >> 16) | (tile0 << 16));
  g1[4] = (int)(tile1 & 0xffffu);  // tile_dim2 = 0
  g1[5] = (int)(uint32_t)(stride0 & 0xffffffffu);
  g1[6] = (int)(uint32_t)((stride0 >> 32) & 0xffffu);  // dim1_stride = 0
  g1[7] = 0;
  const i32x4 z4 = {0, 0, 0, 0};
#if __has_include(<hip/amd_detail/amd_gfx1250_TDM.h>)
  const i32x8 z8 = {0, 0, 0, 0, 0, 0, 0, 0};
  __builtin_amdgcn_tensor_load_to_lds(g0, g1, z4, z4, z8, 0);
#else
  __builtin_amdgcn_tensor_load_to_lds(g0, g1, z4, z4, 0);
#endif
}

// data_size=4B; no pad
#define TDM_F32 0x00020000u
// data_size=4B; pad 4 dwords after every 64 dwords (LDS row stride 68)
#define TDM_F32_PAD64_4 (0x00020000u | (1u << 20) | (5u << 22) | (3u << 25))
// data_size=4B; pad 4 dwords after every 32 dwords (LDS row stride 36)
#define TDM_F32_PAD32_4 (0x00020000u | (1u << 20) | (4u << 22) | (3u << 25))

// LDS byte offset = low 32 bits of the generic (flat) address of a shared var
__device__ __forceinline__ uint32_t lds_addr_of(const void* p) {
  return (uint32_t)(uintptr_t)p;
}

// ---------------------------------------------------------------------------
// Kernel 1: depthwise conv1d over sequence, kernel=6, padding (2,3)
// ---------------------------------------------------------------------------
__global__ __launch_bounds__(256) void dwconv_kernel(
    const float* __restrict__ x, const float* __restrict__ w,
    const float* __restrict__ bias, float* __restrict__ y) {
  const size_t idx = (size_t)blockIdx.x * 256 + threadIdx.x;  // exact cover
  const int c = (int)(idx & (D_ - 1));
  const size_t bs = idx >> 9;  // b*S + s
  const int s = (int)(bs & (S_ - 1));
  const size_t base = (bs - (size_t)s) * D_ + c;
  float acc = bias[c];
#pragma unroll
  for (int k = 0; k < 6; ++k) {
    const int sp = s + k - 2;
    if (sp >= 0 && sp < S_)
      acc = fmaf(x[base + (size_t)sp * D_], w[c * 6 + k], acc);
  }
  y[idx] = acc;
}

// ---------------------------------------------------------------------------
// Kernel 2: C[M,N] = A[M,K] * W[N,K]^T + bias[n] (+ pos[m%S, n] if pos)
// Block tile 128(M) x 64(N); 8 waves of 16(M) x 64(N).
// A tile via TDM (pad -> stride 36). W tile transposed + k-pair interleaved:
// Ws2[kp][n] float2, pitch 80 float2 -> every B fragment is one ds_load_b64,
// lanes 0-15 hit banks 0-31, lanes 16-31 (kp+1, shift 160 dw = bank+32) hit
// banks 32-63: conflict-free.
// If Ct != null also writes C transposed as Ct[b][n][s] (for attention PV).
// ---------------------------------------------------------------------------
__global__ __launch_bounds__(256) void gemm_nt_kernel(
    const float* __restrict__ A, const float* __restrict__ W,
    const float* __restrict__ bias, const float* __restrict__ pos,
    float* __restrict__ C, float* __restrict__ Ct) {
  __shared__ float As[128 * 36];
  __shared__ float Ws[16 * 160];  // [kp 0..15][n 0..79] float2 (n<64 used)
  const int tid = threadIdx.x;
  const int wave = tid >> 5, lane = tid & 31;
  const int l16 = lane & 15, hi = lane >> 4;
  const int m0 = blockIdx.x * 128;
  const int n0 = blockIdx.y * 64;
  const bool leader = tid < 32;
  const uint32_t as_lds = lds_addr_of(As);

  v8f acc[4];
#pragma unroll
  for (int nt = 0; nt < 4; ++nt)
#pragma unroll
    for (int r = 0; r < 8; ++r) acc[nt][r] = 0.0f;

  for (int k0 = 0; k0 < D_; k0 += 32) {
    // DMA A tile 128x32 -> As (row stride 36 via TDM padding)
    if (leader)
      tdm_load_2d(as_lds, A + (size_t)m0 * D_ + k0, /*tile*/ 32, 128,
                  /*tensor*/ 32, 128, /*stride0*/ D_, TDM_F32_PAD32_4);
    // stage W tile transposed, k-pair interleaved: Ws2[kp][n] = {W[n][2kp],
    // W[n][2kp+1]}
    for (int i = tid; i < 64 * 8; i += 256) {
      const int r = i >> 3, c4 = i & 7;
      const float4 v =
          *(const float4*)(W + (size_t)(n0 + r) * D_ + k0 + c4 * 4);
      *(v2f*)&Ws[((2 * c4 + 0) * 80 + r) * 2] = v2f{v.x, v.y};
      *(v2f*)&Ws[((2 * c4 + 1) * 80 + r) * 2] = v2f{v.z, v.w};
    }
    if (leader) __builtin_amdgcn_s_wait_tensorcnt(0);
    __syncthreads();
#pragma unroll
    for (int kc = 0; kc < 8; ++kc) {
      const int kk = kc * 4 + hi * 2;   // lanes 0-15: K 0,1; 16-31: K 2,3
      const int kp = kc * 2 + hi;       // k-pair index
      const v2f af = *(const v2f*)&As[(wave * 16 + l16) * 36 + kk];
#pragma unroll
      for (int nt = 0; nt < 4; ++nt) {
        const v2f bf = *(const v2f*)&Ws[(kp * 80 + nt * 16 + l16) * 2];
        acc[nt] = wmma_f32(af, bf, acc[nt]);
      }
    }
    __syncthreads();
  }

  const int mw = m0 + wave * 16;
#pragma unroll
  for (int nt = 0; nt < 4; ++nt) {
    const int nn = n0 + nt * 16 + l16;
    const float bv = bias[nn];
#pragma unroll
    for (int r = 0; r < 8; ++r) {
      const int mm = mw + hi * 8 + r;
      float v = acc[nt][r] + bv;
      if (pos) v += pos[(size_t)(mm & (S_ - 1)) * D_ + nn];
      C[(size_t)mm * D_ + nn] = v;
      if (Ct) {  // transposed copy: Ct[b][n][s]
        const int bb = mm >> 11, ss = mm & (S_ - 1);
        Ct[((size_t)bb * D_ + nn) * S_ + ss] = v;
      }
    }
  }
}

// ---------------------------------------------------------------------------
// Kernel 3: attention + residual epilogue (two-pass, L2-resident KV).
//   Phase A: Sc[16][2048] = (Q16 @ KV^T)*scale   (WMMA; KV tiles via TDM,
//            KVa[t][k] stride 68 -> b64 fragments, all 64 banks hit once)
//   Phase B: row softmax in LDS
//   Phase C: O^T = KVT_tile @ P^T (WMMA with swapped roles; KT[n][t] pitch 36
//            staged via TDM from the transposed KV copy; both fragments are
//            contiguous b64 loads, Sc fragment shared across 4 n-subtiles)
//   out = O + KV + evo
// LDS: Sc 16x2052 | region = max(Qs 16x516 + KVa 128x68, KT 512x36)
// ---------------------------------------------------------------------------
__global__ __launch_bounds__(256) void attn_kernel(
    const float* __restrict__ Q, const float* __restrict__ KV,
    const float* __restrict__ KVT, const float* __restrict__ evo,
    float* __restrict__ out) {
  extern __shared__ float smem[];
  float* Sc = smem;              // 16 x 2052
  float* Qs = smem + 16 * 2052;  // 16 x 516   (phase A)
  float* KVa = Qs + 16 * 516;    // 128 x 68   (phase A, [t][k], TDM-padded)
  float* KT = Qs;                // 512 x 36   (phase C, [n][t], TDM-padded)

  const int tid = threadIdx.x;
  const int wave = tid >> 5, lane = tid & 31;
  const int l16 = lane & 15, hi = lane >> 4;
  const int batch = blockIdx.x >> 7;
  const int rb = blockIdx.x & 127;
  const size_t qrow0 = (size_t)batch * S_ + rb * 16;
  const size_t kvrow0 = (size_t)batch * S_;
  const size_t kvtbase = (size_t)batch * D_ * S_;
  const bool leader = tid < 32;
  const uint32_t kva_lds = lds_addr_of(KVa);
  const uint32_t kt_lds = lds_addr_of(KT);
  const float scale = 0.044194173824159216f;  // 1/sqrt(512)

  // load 16 Q rows (one-time, cooperative)
  for (int i = tid; i < 16 * 128; i += 256) {
    const int r = i >> 7, c4 = i & 127;
    const float4 v = *(const float4*)(Q + (qrow0 + r) * D_ + c4 * 4);
    *(float4*)(&Qs[r * 516 + c4 * 4]) = v;
  }
  __syncthreads();

  // ---- phase A: scores; wave owns t-tile (tb + wave*16 .. +16)
  for (int tb = 0; tb < S_; tb += 128) {
    v8f acc;
#pragma unroll
    for (int r = 0; r < 8; ++r) acc[r] = 0.0f;
    for (int k0 = 0; k0 < D_; k0 += 64) {
      // DMA KV rows [tb..tb+128) x cols [k0..k0+64) -> KVa (stride 68)
      if (leader) {
        tdm_load_2d(kva_lds, KV + (kvrow0 + tb) * D_ + k0, /*tile*/ 64, 128,
                    /*tensor*/ 64, 128, /*stride0*/ D_, TDM_F32_PAD64_4);
        __builtin_amdgcn_s_wait_tensorcnt(0);
      }
      __syncthreads();
#pragma unroll
      for (int kc = 0; kc < 16; ++kc) {
        const int kk = kc * 4 + hi * 2;
        const v2f af = *(const v2f*)&Qs[l16 * 516 + k0 + kk];
        const v2f bf = *(const v2f*)&KVa[(wave * 16 + l16) * 68 + kk];
        acc = wmma_f32(af, bf, acc);
      }
      __syncthreads();
    }
#pragma unroll
    for (int r = 0; r < 8; ++r)
      Sc[(hi * 8 + r) * 2052 + tb + wave * 16 + l16] = acc[r] * scale;
  }
  __syncthreads();

  // ---- phase B: softmax, 2 rows per wave
#pragma unroll
  for (int rr = 0; rr < 2; ++rr) {
    const int row = wave * 2 + rr;
    float mx = -3.4e38f;
    for (int j = lane; j < S_; j += 32) mx = fmaxf(mx, Sc[row * 2052 + j]);
#pragma unroll
    for (int off = 16; off > 0; off >>= 1)
      mx = fmaxf(mx, __shfl_xor(mx, off, 32));
    float sum = 0.0f;
    for (int j = lane; j < S_; j += 32) {
      const float e = __expf(Sc[row * 2052 + j] - mx);
      Sc[row * 2052 + j] = e;
      sum += e;
    }
#pragma unroll
    for (int off = 16; off > 0; off >>= 1) sum += __shfl_xor(sum, off, 32);
    const float inv = 1.0f / sum;
    for (int j = lane; j < S_; j += 32) Sc[row * 2052 + j] *= inv;
  }
  __syncthreads();

  // ---- phase C: O^T[n][m] = sum_t KVT[n][t] * P[m][t]; wave owns 64 n-rows
  v8f acc2[4];
#pragma unroll
  for (int nt = 0; nt < 4; ++nt)
#pragma unroll
    for (int r = 0; r < 8; ++r) acc2[nt][r] = 0.0f;

  for (int tb = 0; tb < S_; tb += 32) {
    // DMA KVT rows [0..512) x cols [tb..tb+32) -> KT (pitch 36)
    if (leader) {
      tdm_load_2d(kt_lds, KVT + kvtbase + tb, /*tile*/ 32, 512,
                  /*tensor*/ 32, 512, /*stride0*/ S_, TDM_F32_PAD32_4);
      __builtin_amdgcn_s_wait_tensorcnt(0);
    }
    __syncthreads();
#pragma unroll
    for (int kc = 0; kc < 8; ++kc) {
      const int kk = kc * 4 + hi * 2;
      const v2f bf = *(const v2f*)&Sc[l16 * 2052 + tb + kk];  // P^T frag
#pragma unroll
      for (int nt = 0; nt < 4; ++nt) {
        const v2f af =
            *(const v2f*)&KT[(wave * 64 + nt * 16 + l16) * 36 + kk];
        acc2[nt] = wmma_f32(af, bf, acc2[nt]);
      }
    }
    __syncthreads();
  }

  // epilogue: D' lane l16 = m, (hi,r) = n_local; per-lane 8-float runs
#pragma unroll
  for (int nt = 0; nt < 4; ++nt) {
#pragma unroll
    for (int r = 0; r < 8; ++r) {
      const int nn = wave * 64 + nt * 16 + hi * 8 + r;
      const size_t idx = (qrow0 + l16) * D_ + nn;
      out[idx] = acc2[nt][r] + KV[idx] + evo[idx];
    }
  }
}

// ---------------------------------------------------------------------------
extern "C" void kernel_launch(void* const* d_in, const int* in_sizes, int n_in,
                              void* d_out, int out_size, void* d_ws,
                              size_t ws_size, hipStream_t stream) {
  const float* evo = (const float*)d_in[0];  // [8,2048,512]
  const float* Wq = (const float*)d_in[1];   // [512,512]
  const float* bq = (const float*)d_in[2];   // [512]
  const float* dww = (const float*)d_in[3];  // [512,1,6]
  const float* dwb = (const float*)d_in[4];  // [512]
  const float* pww = (const float*)d_in[5];  // [512,512]
  const float* pwb = (const float*)d_in[6];  // [512]
  const float* pos = (const float*)d_in[7];  // [1,2048,512]
  float* outp = (float*)d_out;

  float* conv = (float*)d_ws;              // [M,D]
  float* Qbuf = conv + (size_t)M_ * D_;    // [M,D]
  float* KVbuf = Qbuf + (size_t)M_ * D_;   // [M,D]
  float* KVTbuf = KVbuf + (size_t)M_ * D_; // [B,D,S]

  dwconv_kernel<<<(M_ * D_) / 256, 256, 0, stream>>>(evo, dww, dwb, conv);

  dim3 g(M_ / 128, D_ / 64);
  gemm_nt_kernel<<<g, 256, 0, stream>>>(evo, Wq, bq, pos, Qbuf, nullptr);
  gemm_nt_kernel<<<g, 256, 0, stream>>>(conv, pww, pwb, nullptr, KVbuf,
                                        KVTbuf);

  const size_t shmem = (size_t)(16 * 2052 + 512 * 36) * sizeof(float);
  attn_kernel<<<B_ * (S_ / 16), 256, shmem, stream>>>(Qbuf, KVbuf, KVTbuf,
                                                      evo, outp);
}